// StatementEncoder_88957362635342
// MI455X (gfx1250) — compile-verified
//
#include <hip/hip_runtime.h>
#include <hip/hip_bf16.h>
#include <math.h>

// CDNA5 wave32 fp32 WMMA: D(16x16) = A(16x4) * B(4x16) + C
typedef float v2f __attribute__((ext_vector_type(2)));
typedef float v8f __attribute__((ext_vector_type(8)));

namespace cfg {
constexpr int E  = 512;
constexpr int H  = 8;
constexpr int HD = 64;
constexpr int S  = 512;
constexpr int NB = 64;           // batch
constexpr int WAVES = 8;
constexpr int KSTRIDE = 66;      // padded LDS row stride (floats) for K/V/Q-stage
constexpr int PSTRIDE = 18;      // padded LDS row stride (floats) for P-stage
// dynamic LDS layout in floats
constexpr int OFF_K = 0;
constexpr int OFF_V = OFF_K + S * KSTRIDE;                 // 33792
constexpr int OFF_Q = OFF_V + S * KSTRIDE;                 // 67584
constexpr int OFF_P = OFF_Q + WAVES * 16 * KSTRIDE;        // 76032
constexpr int OFF_R = OFF_P + WAVES * 16 * PSTRIDE;        // 78336
constexpr int SMEM_FLOATS = OFF_R + WAVES * 64;            // 78848 -> 315392 B <= 320KB
}

__device__ __forceinline__ v8f wmma4(v2f a, v2f b, v8f c) {
  // (neg_a, A, neg_b, B, c_mod, C, reuse_a, reuse_b)
  return __builtin_amdgcn_wmma_f32_16x16x4_f32(false, a, false, b, (short)0, c,
                                               false, false);
}

__device__ __forceinline__ v2f ld2(const float* p) { return *(const v2f*)p; }

// Gather one 16x64 tile of x = word_emb[ids] + pos_emb into WMMA A-fragments.
// Fragment j holds columns 4j..4j+3: lane L<16 -> (row L, cols 4j+{0,1}),
// lane L>=16 -> (row L-16, cols 4j+2+{0,1}).
__device__ __forceinline__ void gather_x(v2f xf[16], const int* __restrict__ ids,
                                         const float* __restrict__ wemb,
                                         const float* __restrict__ pemb,
                                         int n, int s0, int emb_off,
                                         int lrow, int lhi) {
  const int row = s0 + lrow;
  const int tok = ids[n * cfg::S + row];
  const float* wrow = wemb + (size_t)tok * cfg::E + emb_off;
  const float* prow = pemb + (size_t)row * cfg::E + emb_off;
#pragma unroll
  for (int j = 0; j < 16; ++j) {
    const int d = 4 * j + 2 * lhi;
    xf[j] = ld2(wrow + d) + ld2(prow + d);
  }
}

// ---------------------------------------------------------------------------
// Kernel 1: Wqv = Wv[1] @ Wq[1]  (64x64, since v = q@Wv^T = x@(Wv@Wq)^T)
// ---------------------------------------------------------------------------
__global__ __launch_bounds__(128) void prep_kernel(const float* __restrict__ Wv1,
                                                   const float* __restrict__ Wq1,
                                                   float* __restrict__ Wqv) {
  const int tid = threadIdx.x;
  const int wave = tid >> 5, lane = tid & 31;
  const int lrow = lane & 15, lhi = lane >> 4;
  const int m0 = wave * 16;
#pragma unroll
  for (int ct = 0; ct < 4; ++ct) {
    const int d0 = ct * 16;
    v8f acc = {};
#pragma unroll
    for (int j = 0; j < 16; ++j) {
      const int k = 4 * j + 2 * lhi;
      v2f af = ld2(Wv1 + (m0 + lrow) * cfg::HD + k);     // A: Wv rows e, contract k
      v2f bf;                                            // B[kidx,d] = Wq[k, d0+d]
      bf.x = Wq1[(k + 0) * cfg::HD + d0 + lrow];
      bf.y = Wq1[(k + 1) * cfg::HD + d0 + lrow];
      acc = wmma4(af, bf, acc);
    }
#pragma unroll
    for (int r = 0; r < 8; ++r)
      Wqv[(m0 + r + 8 * lhi) * cfg::HD + d0 + lrow] = acc[r];
  }
}

// ---------------------------------------------------------------------------
// Kernel 2: one workgroup per (n,h): build K,V in LDS, flash attention with
// fp32 WMMA, masked sequence-sum reduction into cbar[n, h*64 .. h*64+63].
// ---------------------------------------------------------------------------
__global__ __launch_bounds__(256) void attn_kernel(const int* __restrict__ ids,
                                                   const float* __restrict__ amask,
                                                   const float* __restrict__ wemb,
                                                   const float* __restrict__ pemb,
                                                   const float* __restrict__ Wq1,
                                                   const float* __restrict__ Wk1,
                                                   const float* __restrict__ Wqv,
                                                   float* __restrict__ cbar) {
  extern __shared__ float sm[];
  float* Ksm = sm + cfg::OFF_K;
  float* Vsm = sm + cfg::OFF_V;
  float* Qst = sm + cfg::OFF_Q;
  float* Pst = sm + cfg::OFF_P;
  float* red = sm + cfg::OFF_R;

  const int n = blockIdx.x / cfg::H;
  const int h = blockIdx.x % cfg::H;
  const int tid = threadIdx.x;
  const int wave = tid >> 5, lane = tid & 31;
  const int lrow = lane & 15, lhi = lane >> 4;
  const int emb_off = h * cfg::HD;

  float* QstW = Qst + wave * 16 * cfg::KSTRIDE;
  float* PstW = Pst + wave * 16 * cfg::PSTRIDE;

  // ---- Phase 1: K = x@Wk^T, V = x@Wqv^T  -> LDS --------------------------
  for (int t = 0; t < 4; ++t) {
    const int s0 = (wave + 8 * t) * 16;
    v2f xf[16];
    gather_x(xf, ids, wemb, pemb, n, s0, emb_off, lrow, lhi);
#pragma unroll
    for (int e0t = 0; e0t < 4; ++e0t) {
      const int e0 = e0t * 16;
      v8f ak = {}, av = {};
#pragma unroll
      for (int j = 0; j < 16; ++j) {
        const int d = 4 * j + 2 * lhi;
        v2f bk = ld2(Wk1 + (e0 + lrow) * cfg::HD + d);   // B[d,e]=Wk[e,d]
        v2f bv = ld2(Wqv + (e0 + lrow) * cfg::HD + d);
        ak = wmma4(xf[j], bk, ak);
        av = wmma4(xf[j], bv, av);
      }
#pragma unroll
      for (int r = 0; r < 8; ++r) {
        const int srow = s0 + r + 8 * lhi;
        Ksm[srow * cfg::KSTRIDE + e0 + lrow] = ak[r];
        Vsm[srow * cfg::KSTRIDE + e0 + lrow] = av[r];
      }
    }
  }
  __syncthreads();

  const float rs = 0.044194173824159216f;  // 1/sqrt(E)
  float colAcc[4] = {0.f, 0.f, 0.f, 0.f};

  // ---- Phase 2: per-wave query tiles, online softmax ---------------------
  for (int t = 0; t < 4; ++t) {
    const int s0 = (wave + 8 * t) * 16;
    // Q tile = x@Wq^T: compute (C-layout), stage to LDS, re-read as A-frags
    {
      v2f xf[16];
      gather_x(xf, ids, wemb, pemb, n, s0, emb_off, lrow, lhi);
#pragma unroll
      for (int e0t = 0; e0t < 4; ++e0t) {
        const int e0 = e0t * 16;
        v8f qa = {};
#pragma unroll
        for (int j = 0; j < 16; ++j) {
          const int d = 4 * j + 2 * lhi;
          v2f bq = ld2(Wq1 + (e0 + lrow) * cfg::HD + d);
          qa = wmma4(xf[j], bq, qa);
        }
#pragma unroll
        for (int r = 0; r < 8; ++r)
          QstW[(r + 8 * lhi) * cfg::KSTRIDE + e0 + lrow] = qa[r];
      }
    }
    asm volatile("s_wait_dscnt 0" ::: "memory");
    v2f qf[16];
#pragma unroll
    for (int j = 0; j < 16; ++j)
      qf[j] = ld2(&QstW[lrow * cfg::KSTRIDE + 4 * j + 2 * lhi]);

    float M[8], Ssum[8];
#pragma unroll
    for (int r = 0; r < 8; ++r) { M[r] = -INFINITY; Ssum[r] = 0.f; }
    v8f ctx[4];
#pragma unroll
    for (int e0t = 0; e0t < 4; ++e0t) ctx[e0t] = (v8f){};

    for (int kt = 0; kt < cfg::S / 16; ++kt) {
      const int k0 = kt * 16;
      // energy tile = Q(16x64) @ K^T
      v8f e8 = {};
#pragma unroll
      for (int j = 0; j < 16; ++j) {
        const int d = 4 * j + 2 * lhi;
        v2f bf = ld2(&Ksm[(k0 + lrow) * cfg::KSTRIDE + d]);
        e8 = wmma4(qf[j], bf, e8);
      }
      const float mk = amask[n * cfg::S + k0 + lrow];  // key-column mask
      float p[8], corr[8];
#pragma unroll
      for (int r = 0; r < 8; ++r)
        p[r] = ((mk == 0.f) ? -1e20f : e8[r]) * rs;
      // row-max over the 16 key columns (lanes), online update
#pragma unroll
      for (int r = 0; r < 8; ++r) {
        float m = p[r];
        m = fmaxf(m, __shfl_xor(m, 1));
        m = fmaxf(m, __shfl_xor(m, 2));
        m = fmaxf(m, __shfl_xor(m, 4));
        m = fmaxf(m, __shfl_xor(m, 8));
        const float nm = fmaxf(M[r], m);
        corr[r] = expf(M[r] - nm);
        M[r] = nm;
        p[r] = expf(p[r] - nm);
      }
#pragma unroll
      for (int r = 0; r < 8; ++r) {
        float s = p[r];
        s += __shfl_xor(s, 1);
        s += __shfl_xor(s, 2);
        s += __shfl_xor(s, 4);
        s += __shfl_xor(s, 8);
        Ssum[r] = Ssum[r] * corr[r] + s;
      }
#pragma unroll
      for (int e0t = 0; e0t < 4; ++e0t)
#pragma unroll
        for (int r = 0; r < 8; ++r) ctx[e0t][r] *= corr[r];
      // stage P (C-layout -> A-layout) and accumulate ctx += P @ V
#pragma unroll
      for (int r = 0; r < 8; ++r)
        PstW[(r + 8 * lhi) * cfg::PSTRIDE + lrow] = p[r];
      asm volatile("s_wait_dscnt 0" ::: "memory");
      v2f pf[4];
#pragma unroll
      for (int c = 0; c < 4; ++c)
        pf[c] = ld2(&PstW[lrow * cfg::PSTRIDE + 4 * c + 2 * lhi]);
#pragma unroll
      for (int e0t = 0; e0t < 4; ++e0t) {
        const int e0 = e0t * 16;
#pragma unroll
        for (int c = 0; c < 4; ++c) {
          const int kr = k0 + 4 * c + 2 * lhi;
          v2f bf;
          bf.x = Vsm[(kr + 0) * cfg::KSTRIDE + e0 + lrow];
          bf.y = Vsm[(kr + 1) * cfg::KSTRIDE + e0 + lrow];
          ctx[e0t] = wmma4(pf[c], bf, ctx[e0t]);
        }
      }
    }

    // normalize, apply query-row mask, reduce rows of this tile
#pragma unroll
    for (int e0t = 0; e0t < 4; ++e0t) {
      float partial = 0.f;
#pragma unroll
      for (int r = 0; r < 8; ++r) {
        const int row = s0 + r + 8 * lhi;
        partial += (ctx[e0t][r] / Ssum[r]) * amask[n * cfg::S + row];
      }
      partial += __shfl_xor(partial, 16);  // merge the two row-groups
      colAcc[e0t] += partial;
    }
  }

  // ---- cross-wave reduction into cbar ------------------------------------
  if (lane < 16) {
#pragma unroll
    for (int e0t = 0; e0t < 4; ++e0t)
      red[wave * 64 + e0t * 16 + lane] = colAcc[e0t];
  }
  __syncthreads();
  if (tid < 64) {
    float s = 0.f;
#pragma unroll
    for (int w = 0; w < cfg::WAVES; ++w) s += red[w * 64 + tid];
    cbar[n * cfg::E + emb_off + tid] = s;
  }
}

// ---------------------------------------------------------------------------
// Kernel 3: out[n,e] = (cbar[n,:]·Wo[e,:] + bo[e]*msum[n]) / S
// ---------------------------------------------------------------------------
__global__ __launch_bounds__(256) void outproj_kernel(const float* __restrict__ cbar,
                                                      const float* __restrict__ Wo1,
                                                      const float* __restrict__ bo1,
                                                      const float* __restrict__ amask,
                                                      float* __restrict__ out) {
  __shared__ float msum[cfg::NB];
  const int tid = threadIdx.x;
  if (tid < cfg::NB) {
    float s = 0.f;
    for (int j = 0; j < cfg::S; ++j) s += amask[tid * cfg::S + j];
    msum[tid] = s;
  }
  __syncthreads();
  const int wave = tid >> 5, lane = tid & 31;
  const int lrow = lane & 15, lhi = lane >> 4;
  for (int idx = wave; idx < 128; idx += 8) {
    const int n0 = (idx & 3) * 16;
    const int e0 = (idx >> 2) * 16;
    v8f acc = {};
#pragma unroll 8
    for (int j = 0; j < cfg::E / 4; ++j) {
      const int c = 4 * j + 2 * lhi;
      v2f af = ld2(cbar + (n0 + lrow) * cfg::E + c);
      v2f bf = ld2(Wo1 + (e0 + lrow) * cfg::E + c);  // B[c,e]=Wo[e,c]
      acc = wmma4(af, bf, acc);
    }
#pragma unroll
    for (int r = 0; r < 8; ++r) {
      const int n = n0 + r + 8 * lhi;
      const int e = e0 + lrow;
      out[n * cfg::E + e] = (acc[r] + bo1[e] * msum[n]) * (1.0f / (float)cfg::S);
    }
  }
}

// ---------------------------------------------------------------------------
extern "C" void kernel_launch(void* const* d_in, const int* in_sizes, int n_in,
                              void* d_out, int out_size, void* d_ws, size_t ws_size,
                              hipStream_t stream) {
  (void)in_sizes; (void)n_in; (void)out_size; (void)ws_size;
  const int*   ids  = (const int*)d_in[0];
  const float* mask = (const float*)d_in[1];
  const float* wemb = (const float*)d_in[2];
  const float* pemb = (const float*)d_in[3];
  const float* Wq   = (const float*)d_in[4];
  const float* Wk   = (const float*)d_in[5];
  const float* Wv   = (const float*)d_in[6];
  const float* Wo   = (const float*)d_in[7];
  const float* bo   = (const float*)d_in[8];
  float* out = (float*)d_out;

  // Only the last layer (l = L-1 = 1) affects the output.
  const float* Wq1 = Wq + cfg::HD * cfg::HD;
  const float* Wk1 = Wk + cfg::HD * cfg::HD;
  const float* Wv1 = Wv + cfg::HD * cfg::HD;
  const float* Wo1 = Wo + cfg::E * cfg::E;
  const float* bo1 = bo + cfg::E;

  float* Wqv  = (float*)d_ws;                 // 64x64
  float* cbar = Wqv + cfg::HD * cfg::HD;      // 64x512

  prep_kernel<<<1, 128, 0, stream>>>(Wv1, Wq1, Wqv);

  const size_t smem = (size_t)cfg::SMEM_FLOATS * sizeof(float);  // ~308 KB
  hipFuncSetAttribute((const void*)attn_kernel,
                      hipFuncAttributeMaxDynamicSharedMemorySize, (int)smem);
  attn_kernel<<<dim3(cfg::NB * cfg::H), 256, smem, stream>>>(
      ids, mask, wemb, pemb, Wq1, Wk1, Wqv, cbar);

  outproj_kernel<<<1, 256, 0, stream>>>(cbar, Wo1, bo1, mask, out);
}